// HyperGAP_50972671869731
// MI455X (gfx1250) — compile-verified
//
#include <hip/hip_runtime.h>
#include <hip/hip_bf16.h>

#define N_NODES 100000
#define N_EDGES 200000
#define NNZ     800000
#define D       128
#define P       32

typedef __attribute__((ext_vector_type(2))) float v2f;
typedef __attribute__((ext_vector_type(8))) float v8f;

// ---------------------------------------------------------------------------
// Segment counts: one thread per pin, float atomics (counts reused as means)
// ---------------------------------------------------------------------------
__global__ void count_kernel(const int* __restrict__ nidx,
                             const int* __restrict__ eidx,
                             float* __restrict__ ecnt,
                             float* __restrict__ ncnt) {
    int t = blockIdx.x * blockDim.x + threadIdx.x;
    if (t < NNZ) {
        atomicAdd(&ecnt[eidx[t]], 1.0f);
        atomicAdd(&ncnt[nidx[t]], 1.0f);
    }
}

__global__ void recip_kernel(float* __restrict__ c, int n) {
    int t = blockIdx.x * blockDim.x + threadIdx.x;
    if (t < n) c[t] = 1.0f / fmaxf(c[t], 1.0f);
}

// ---------------------------------------------------------------------------
// Scatter-add: dst[didx[p]][:] += src[sidx[p]][:]
// One wave per pin: 32 lanes x float4 = 128 floats (512B coalesced gather,
// contiguous global_atomic_add_f32 scatter; no intra-wave address conflicts).
// ---------------------------------------------------------------------------
__global__ void scatter_kernel(const float* __restrict__ src,
                               const int*   __restrict__ sidx,
                               const int*   __restrict__ didx,
                               float*       __restrict__ dst) {
    int t   = blockIdx.x * blockDim.x + threadIdx.x;
    int pin = t >> 5;
    int c4  = (t & 31) << 2;
    if (pin < NNZ) {
        const float4 v = *(const float4*)&src[(size_t)sidx[pin] * D + c4];
        float* d = &dst[(size_t)didx[pin] * D + c4];
        atomicAdd(d + 0, v.x);
        atomicAdd(d + 1, v.y);
        atomicAdd(d + 2, v.z);
        atomicAdd(d + 3, v.w);
    }
}

// Row scale by reciprocal count: one wave per row, float4 per lane
__global__ void scale_kernel(float* __restrict__ feat,
                             const float* __restrict__ inv,
                             int nrows) {
    int t   = blockIdx.x * blockDim.x + threadIdx.x;
    int row = t >> 5;
    int c4  = (t & 31) << 2;
    if (row < nrows) {
        float s = inv[row];
        float4* p = (float4*)&feat[(size_t)row * D + c4];
        float4 v = *p;
        v.x *= s; v.y *= s; v.z *= s; v.w *= s;
        *p = v;
    }
}

// ---------------------------------------------------------------------------
// GEMM: Y(N x DOUT) = act(X(N x 128) @ W(128 x DOUT) + bias)
// 32 rows per block. A tile in LDS (pitch 132), W staged TRANSPOSED into LDS
// in K-slabs (pitch KSLAB+4): both pitches are 4 mod 64 banks, so the 16-lane
// float2 fragment reads are bank-conflict-free ds_load_b64.
// DOUT=128: 256 thr / 8 waves, 2 independent accumulators per wave (two WMMA
// dep chains sharing one A fragment). DOUT=32: 128 thr / 4 waves, 1 acc each.
// ---------------------------------------------------------------------------
template <int DOUT>
__global__ __launch_bounds__(DOUT == 128 ? 256 : 128)
void gemm_kernel(const float* __restrict__ X,
                 const float* __restrict__ W,
                 const float* __restrict__ bias,
                 float*       __restrict__ Y,
                 int do_relu) {
    constexpr int BLOCK  = (DOUT == 128) ? 256 : 128;
    constexpr int ROWS   = 32;
    constexpr int KSLAB  = (DOUT == 128) ? 64 : 128;
    constexpr int WPITCH = KSLAB + 4;   // 68 or 132; mod 64 == 4
    constexpr int APITCH = 132;

    __shared__ float As[ROWS * APITCH];
    __shared__ float Ws[DOUT * WPITCH];

    const int tid      = threadIdx.x;
    const int blockrow = blockIdx.x * ROWS;

    // ---- stage A tile: 32 rows x 128 cols, coalesced float4 ----
#pragma unroll
    for (int i = 0; i < (ROWS * D / 4) / BLOCK; ++i) {
        int idx = i * BLOCK + tid;
        int r   = idx >> 5;
        int c4  = (idx & 31) << 2;
        *(float4*)&As[r * APITCH + c4] =
            *(const float4*)&X[(size_t)(blockrow + r) * D + c4];
    }

    const int wave = tid >> 5;
    const int lane = tid & 31;
    const int kh   = lane >> 4;   // K-half (A,B) / M-half (C,D)
    const int mn   = lane & 15;   // M for A, N for B/C/D

    v8f acc0 = {}, acc1 = {};

    for (int kbase = 0; kbase < D; kbase += KSLAB) {
        __syncthreads();   // previous slab fully consumed / A staged
        // ---- stage W slab transposed: Ws[col][k_local] ----
#pragma unroll
        for (int i = 0; i < (KSLAB * DOUT / 4) / BLOCK; ++i) {
            int idx = i * BLOCK + tid;
            int kl  = idx / (DOUT / 4);
            int c4  = (idx % (DOUT / 4)) << 2;
            float4 v = *(const float4*)&W[(size_t)(kbase + kl) * DOUT + c4];
            Ws[(c4 + 0) * WPITCH + kl] = v.x;
            Ws[(c4 + 1) * WPITCH + kl] = v.y;
            Ws[(c4 + 2) * WPITCH + kl] = v.z;
            Ws[(c4 + 3) * WPITCH + kl] = v.w;
        }
        __syncthreads();

        if constexpr (DOUT == 128) {
            const int rt  = wave >> 2;            // row tile 0/1
            const int ct0 = (wave & 3) * 2;       // col tiles ct0, ct0+1
            const float* ar = &As[(rt * 16 + mn) * APITCH + kbase];
            const float* b0 = &Ws[(ct0 * 16 + mn) * WPITCH];
            const float* b1 = &Ws[(ct0 * 16 + 16 + mn) * WPITCH];
#pragma unroll
            for (int kb = 0; kb < KSLAB; kb += 4) {
                v2f a  = *(const v2f*)&ar[kb + 2 * kh];
                v2f f0 = *(const v2f*)&b0[kb + 2 * kh];
                v2f f1 = *(const v2f*)&b1[kb + 2 * kh];
                acc0 = __builtin_amdgcn_wmma_f32_16x16x4_f32(
                    false, a, false, f0, (short)0, acc0, false, false);
                acc1 = __builtin_amdgcn_wmma_f32_16x16x4_f32(
                    false, a, false, f1, (short)0, acc1, false, false);
            }
        } else {
            const int rt = wave >> 1;
            const int ct = wave & 1;
            const float* ar = &As[(rt * 16 + mn) * APITCH + kbase];
            const float* b0 = &Ws[(ct * 16 + mn) * WPITCH];
#pragma unroll
            for (int kb = 0; kb < KSLAB; kb += 4) {
                v2f a  = *(const v2f*)&ar[kb + 2 * kh];
                v2f f0 = *(const v2f*)&b0[kb + 2 * kh];
                acc0 = __builtin_amdgcn_wmma_f32_16x16x4_f32(
                    false, a, false, f0, (short)0, acc0, false, false);
            }
        }
    }

    // ---- epilogue: bias + optional ReLU, C/D layout M=i (+8 upper half) ----
    const int rt  = (DOUT == 128) ? (wave >> 2) : (wave >> 1);
    const int rowbase = blockrow + rt * 16;
    {
        const int ct  = (DOUT == 128) ? ((wave & 3) * 2) : (wave & 1);
        const int col = ct * 16 + mn;
        float bv = bias[col];
#pragma unroll
        for (int i = 0; i < 8; ++i) {
            float v = acc0[i] + bv;
            if (do_relu) v = fmaxf(v, 0.0f);
            Y[(size_t)(rowbase + i + 8 * kh) * DOUT + col] = v;
        }
    }
    if constexpr (DOUT == 128) {
        const int col = ((wave & 3) * 2 + 1) * 16 + mn;
        float bv = bias[col];
#pragma unroll
        for (int i = 0; i < 8; ++i) {
            float v = acc1[i] + bv;
            if (do_relu) v = fmaxf(v, 0.0f);
            Y[(size_t)(rowbase + i + 8 * kh) * DOUT + col] = v;
        }
    }
}

// ---------------------------------------------------------------------------
// Softmax over P=32: one wave32 per row, shfl_xor tree reductions
// ---------------------------------------------------------------------------
__global__ void softmax_kernel(const float* __restrict__ logits,
                               float* __restrict__ out) {
    int t    = blockIdx.x * blockDim.x + threadIdx.x;
    int row  = t >> 5;
    int lane = t & 31;
    if (row < N_NODES) {
        float v = logits[(size_t)row * P + lane];
        float m = v;
#pragma unroll
        for (int o = 16; o > 0; o >>= 1) m = fmaxf(m, __shfl_xor(m, o, 32));
        float e = __expf(v - m);
        float s = e;
#pragma unroll
        for (int o = 16; o > 0; o >>= 1) s += __shfl_xor(s, o, 32);
        out[(size_t)row * P + lane] = e / s;
    }
}

// ---------------------------------------------------------------------------
extern "C" void kernel_launch(void* const* d_in, const int* in_sizes, int n_in,
                              void* d_out, int out_size, void* d_ws, size_t ws_size,
                              hipStream_t stream) {
    const float* x    = (const float*)d_in[0];
    const int*   nidx = (const int*)  d_in[1];
    const int*   eidx = (const int*)  d_in[2];
    const float* W1   = (const float*)d_in[3];
    const float* b1   = (const float*)d_in[4];
    const float* W2   = (const float*)d_in[5];
    const float* b2   = (const float*)d_in[6];
    const float* Wm1  = (const float*)d_in[7];
    const float* bm1  = (const float*)d_in[8];
    const float* Wm2  = (const float*)d_in[9];
    const float* bm2  = (const float*)d_in[10];
    float* outp = (float*)d_out;

    // Workspace layout (floats), aliased across phases:
    float* ws    = (float*)d_ws;
    float* ecnt  = ws;                                   // E
    float* ncnt  = ws + N_EDGES;                         // N
    float* efeat = ws + N_EDGES + N_NODES;               // E*D (102.4 MB)
    float* agg   = efeat + (size_t)N_EDGES * D;          // N*D (51.2 MB)
    float* hA    = agg   + (size_t)N_NODES * D;          // N*D (51.2 MB)
    float* hB     = efeat;   // hconv2 output reuses efeat region
    float* h3     = hA;      // MLP hidden reuses hA
    float* logits = agg;     // logits reuse agg

    const int BLK = 256;
    const int pinGrid  = (NNZ * 32) / BLK;       // 100000
    const int eRowGrid = (N_EDGES * 32) / BLK;   // 25000
    const int nRowGrid = (N_NODES * 32) / BLK;   // 12500
    const int gemmGrid = N_NODES / 32;           // 3125

    // --- segment counts -> reciprocals (shared by both hconvs) ---
    hipMemsetAsync(ecnt, 0, (size_t)(N_EDGES + N_NODES) * sizeof(float), stream);
    count_kernel<<<(NNZ + BLK - 1) / BLK, BLK, 0, stream>>>(nidx, eidx, ecnt, ncnt);
    recip_kernel<<<((N_EDGES + N_NODES) + BLK - 1) / BLK, BLK, 0, stream>>>(
        ecnt, N_EDGES + N_NODES);

    // --- hconv1: x -> hA ---
    hipMemsetAsync(efeat, 0, (size_t)N_EDGES * D * sizeof(float), stream);
    scatter_kernel<<<pinGrid, BLK, 0, stream>>>(x, nidx, eidx, efeat);
    scale_kernel<<<eRowGrid, BLK, 0, stream>>>(efeat, ecnt, N_EDGES);
    hipMemsetAsync(agg, 0, (size_t)N_NODES * D * sizeof(float), stream);
    scatter_kernel<<<pinGrid, BLK, 0, stream>>>(efeat, eidx, nidx, agg);
    scale_kernel<<<nRowGrid, BLK, 0, stream>>>(agg, ncnt, N_NODES);
    gemm_kernel<128><<<gemmGrid, 256, 0, stream>>>(agg, W1, b1, hA, 1);

    // --- hconv2: hA -> hB (aliases efeat; efeat fully consumed before GEMM) ---
    hipMemsetAsync(efeat, 0, (size_t)N_EDGES * D * sizeof(float), stream);
    scatter_kernel<<<pinGrid, BLK, 0, stream>>>(hA, nidx, eidx, efeat);
    scale_kernel<<<eRowGrid, BLK, 0, stream>>>(efeat, ecnt, N_EDGES);
    hipMemsetAsync(agg, 0, (size_t)N_NODES * D * sizeof(float), stream);
    scatter_kernel<<<pinGrid, BLK, 0, stream>>>(efeat, eidx, nidx, agg);
    scale_kernel<<<nRowGrid, BLK, 0, stream>>>(agg, ncnt, N_NODES);
    gemm_kernel<128><<<gemmGrid, 256, 0, stream>>>(agg, W2, b2, hB, 1);

    // --- MLP hidden: h3 = relu(hB @ Wm1 + bm1) ---
    gemm_kernel<128><<<gemmGrid, 256, 0, stream>>>(hB, Wm1, bm1, h3, 1);

    // --- logits = h3 @ Wm2 + bm2 ; softmax -> out ---
    gemm_kernel<32><<<gemmGrid, 128, 0, stream>>>(h3, Wm2, bm2, logits, 0);
    softmax_kernel<<<nRowGrid, BLK, 0, stream>>>(logits, outp);
}